// GATv2Convolution_3324304687446
// MI455X (gfx1250) — compile-verified
//
#include <hip/hip_runtime.h>
#include <math.h>

#define HC        64
#define HEADS     8
#define LEAKY     0.2f
#define MT        2            // M-tiles (of 16 edges) per wave -> 32 edges/wave
#define WPB       2            // waves per block -> 64 edges/block
#define ROWS      132          // padded LDS row stride (128+4): float4-aligned, conflict-free

typedef __attribute__((ext_vector_type(2))) float v2f;
typedef __attribute__((ext_vector_type(8))) float v8f;

// Monotone float <-> uint mapping so atomicMax(uint) == float max
__device__ __forceinline__ unsigned f2o(float f) {
    unsigned u = __float_as_uint(f);
    return (u & 0x80000000u) ? ~u : (u | 0x80000000u);
}
__device__ __forceinline__ float o2f(unsigned u) {
    return (u & 0x80000000u) ? __uint_as_float(u & 0x7FFFFFFFu) : __uint_as_float(~u);
}

// ---------------------------------------------------------------------------
// Init: m = ord(-inf), den = 0, out = 0, and build quad-packed w:
//   wq[(hi*16 + kb/8)*64 + n] = { w[k][n], w[k+1][n], w[k+4][n], w[k+5][n] },
//   k = 8*(kb/8) + 2*hi  -> one b128 load supplies B fragments for 2 k-steps.
// ---------------------------------------------------------------------------
__global__ void gat_init_kernel(unsigned* __restrict__ m_ord, float* __restrict__ den,
                                float* __restrict__ out, float4* __restrict__ wq,
                                const float* __restrict__ w,
                                int nm, int nout, int nquads) {
    int t = blockIdx.x * blockDim.x + threadIdx.x;
    if (t < nm) { m_ord[t] = f2o(-INFINITY); den[t] = 0.0f; }
    if (t < nout) out[t] = 0.0f;
    if (t < nquads) {                                  // nquads = 2*16*64 = 2048
        const int hi = t >> 10;
        const int q8 = (t >> 6) & 15;
        const int n  = t & 63;
        const int k  = 8 * q8 + 2 * hi;
        wq[t] = make_float4(w[(size_t)k * HC + n],       w[(size_t)(k + 1) * HC + n],
                            w[(size_t)(k + 4) * HC + n], w[(size_t)(k + 5) * HC + n]);
    }
}

// ---------------------------------------------------------------------------
// Pass 1: per-edge scores via WMMA f32 GEMM (32 edges x 64 out, K=128) per wave.
// Quad-packed B: one global_load_b128 feeds 2 k-steps x (reused over 2 M-tiles).
// Fused leaky_relu * a + per-head channel reduce; atomicMax into m_ord.
// ---------------------------------------------------------------------------
__global__ void __launch_bounds__(32 * WPB)
gat_edge_score_kernel(const float* __restrict__ h, const int* __restrict__ src,
                      const int* __restrict__ dst, const float4* __restrict__ wq,
                      const float* __restrict__ a, float* __restrict__ scores,
                      unsigned* __restrict__ m_ord, int n_edges) {
    __shared__ float lds[WPB * MT * 16 * ROWS];        // 64 rows x 132 floats = 33792 B

    const int wave  = threadIdx.x >> 5;
    const int lane  = threadIdx.x & 31;
    const int ebase = blockIdx.x * (WPB * MT * 16) + wave * (MT * 16);
    const int rbase = wave * (MT * 16);                // this wave's LDS row base

    // ---- Stage hij tile (32 edges x 128 f32) into this wave's LDS rows ----
#pragma unroll
    for (int j = 0; j < 2; ++j) {
        const int task = lane + 32 * j;                // 64 (row, half) tasks
        const int row  = task & 31;
        const int half = task >> 5;                    // 0: hi = h[src], 1: hj = h[dst]
        int eg = ebase + row;
        if (eg >= n_edges) eg = n_edges - 1;           // clamp (tail-safe)
        const int idx = half ? dst[eg] : src[eg];
        const float4* sv = (const float4*)(h + (size_t)idx * HC);
        float4* rv = (float4*)&lds[(rbase + row) * ROWS + half * HC];
#pragma unroll
        for (int q = 0; q < HC / 4; ++q) rv[q] = sv[q];
    }
    __syncthreads();

    // ---- WMMA f32 16x16x4, K = 128, 2 M-tiles x 4 N-tiles ----
    const int mrow = lane & 15;
    const int hi   = lane >> 4;
    const int hi2  = hi * 2;
    const int ncol = lane & 15;

    v8f acc[MT][4];
#pragma unroll
    for (int mt = 0; mt < MT; ++mt)
#pragma unroll
        for (int nt = 0; nt < 4; ++nt)
            acc[mt][nt] = (v8f){0.f,0.f,0.f,0.f,0.f,0.f,0.f,0.f};

#pragma unroll
    for (int kb8 = 0; kb8 < 16; ++kb8) {               // each iter covers k-steps kb, kb+4
        const int kb = kb8 * 8;
        float4 bq[4];
        const float4* wrow = wq + ((size_t)hi * 16 + kb8) * HC;
#pragma unroll
        for (int nt = 0; nt < 4; ++nt) bq[nt] = wrow[nt * 16 + ncol];

#pragma unroll
        for (int mt = 0; mt < MT; ++mt) {
            const float* arow = &lds[(rbase + mt * 16 + mrow) * ROWS];
            v2f a0, a1;
            a0.x = arow[kb + hi2];     a0.y = arow[kb + hi2 + 1];
            a1.x = arow[kb + 4 + hi2]; a1.y = arow[kb + 4 + hi2 + 1];
#pragma unroll
            for (int nt = 0; nt < 4; ++nt) {
                v2f b0; b0.x = bq[nt].x; b0.y = bq[nt].y;
                acc[mt][nt] = __builtin_amdgcn_wmma_f32_16x16x4_f32(
                    false, a0, false, b0, (short)0, acc[mt][nt], false, false);
            }
#pragma unroll
            for (int nt = 0; nt < 4; ++nt) {
                v2f b1; b1.x = bq[nt].z; b1.y = bq[nt].w;
                acc[mt][nt] = __builtin_amdgcn_wmma_f32_16x16x4_f32(
                    false, a1, false, b1, (short)0, acc[mt][nt], false, false);
            }
        }
    }

    // ---- Epilogue: leaky_relu, *a, per-head reduce (8 lanes) via shuffles ----
    const int headInNt = (lane >> 3) & 1;
#pragma unroll
    for (int nt = 0; nt < 4; ++nt) {
        const float av = a[nt * 16 + ncol];
#pragma unroll
        for (int mt = 0; mt < MT; ++mt) {
#pragma unroll
            for (int i = 0; i < 8; ++i) {
                float v = acc[mt][nt][i];
                v = (v > 0.0f) ? v : (LEAKY * v);
                v *= av;
                v += __shfl_xor(v, 1, 32);
                v += __shfl_xor(v, 2, 32);
                v += __shfl_xor(v, 4, 32);
                if ((lane & 7) == 0) {                 // lanes 0,8,16,24
                    const int edge = ebase + mt * 16 + i + 8 * hi;
                    if (edge < n_edges) {
                        const int head = nt * 2 + headInNt;
                        scores[(size_t)edge * HEADS + head] = v;
                        atomicMax(&m_ord[(size_t)dst[edge] * HEADS + head], f2o(v));
                    }
                }
            }
        }
    }
}

// ---------------------------------------------------------------------------
// Pass 2: den[dst,head] += exp(score - m[dst,head])
// ---------------------------------------------------------------------------
__global__ void gat_den_kernel(const float* __restrict__ scores, const int* __restrict__ dst,
                               const unsigned* __restrict__ m_ord, float* __restrict__ den,
                               int total) {
    int t = blockIdx.x * blockDim.x + threadIdx.x;
    if (t >= total) return;
    const int e = t >> 3, hd = t & 7;
    const int d = dst[e];
    const float m = o2f(m_ord[(size_t)d * HEADS + hd]);
    atomicAdd(&den[(size_t)d * HEADS + hd], __expf(scores[t] - m));
}

// ---------------------------------------------------------------------------
// Pass 3: one thread per (edge, head): attn computed once, 8-channel scatter.
// ---------------------------------------------------------------------------
__global__ void gat_scatter_kernel(const float* __restrict__ h, const int* __restrict__ src,
                                   const int* __restrict__ dst, const float* __restrict__ scores,
                                   const unsigned* __restrict__ m_ord, const float* __restrict__ den,
                                   float* __restrict__ out, int total) {
    int t = blockIdx.x * blockDim.x + threadIdx.x;
    if (t >= total) return;                            // total = E * HEADS
    const int e  = t >> 3;
    const int hd = t & 7;
    const int d  = dst[e];
    const int s  = src[e];
    const float m    = o2f(m_ord[(size_t)d * HEADS + hd]);
    const float attn = __expf(scores[t] - m) / den[(size_t)d * HEADS + hd];
    const float4* hv = (const float4*)(h + (size_t)s * HC + hd * 8);
    const float4 v0 = hv[0];
    const float4 v1 = hv[1];
    float* ob = out + (size_t)d * HC + hd * 8;
    atomicAdd(ob + 0, v0.x * attn);
    atomicAdd(ob + 1, v0.y * attn);
    atomicAdd(ob + 2, v0.z * attn);
    atomicAdd(ob + 3, v0.w * attn);
    atomicAdd(ob + 4, v1.x * attn);
    atomicAdd(ob + 5, v1.y * attn);
    atomicAdd(ob + 6, v1.z * attn);
    atomicAdd(ob + 7, v1.w * attn);
}

// ---------------------------------------------------------------------------
extern "C" void kernel_launch(void* const* d_in, const int* in_sizes, int n_in,
                              void* d_out, int out_size, void* d_ws, size_t ws_size,
                              hipStream_t stream) {
    const float* h   = (const float*)d_in[0];
    const int*   src = (const int*)d_in[1];
    const int*   dst = (const int*)d_in[2];
    const float* w   = (const float*)d_in[3];
    const float* a   = (const float*)d_in[4];
    float* out = (float*)d_out;

    const int n_edges = in_sizes[1];
    const int n_nodes = in_sizes[0] / HC;

    // Workspace: scores (E*8 f32) | m_ord (N*8 u32) | den (N*8 f32) | wq (2048 float4)
    char* ws = (char*)d_ws;
    size_t off = 0;
    float*    scores = (float*)(ws + off);    off += (size_t)n_edges * HEADS * sizeof(float);
    unsigned* m_ord  = (unsigned*)(ws + off); off += (size_t)n_nodes * HEADS * sizeof(unsigned);
    float*    den    = (float*)(ws + off);    off += (size_t)n_nodes * HEADS * sizeof(float);
    float4*   wq     = (float4*)(ws + off);

    const int nm     = n_nodes * HEADS;
    const int nout   = n_nodes * HC;
    const int nquads = 2 * 16 * HC;                    // 2048
    int initN = nout > nm ? nout : nm;
    if (nquads > initN) initN = nquads;
    gat_init_kernel<<<(initN + 255) / 256, 256, 0, stream>>>(m_ord, den, out, wq, w,
                                                             nm, nout, nquads);

    const int epb    = WPB * MT * 16;                  // 64 edges per block (2 waves)
    const int blocks = (n_edges + epb - 1) / epb;
    gat_edge_score_kernel<<<blocks, 32 * WPB, 0, stream>>>(h, src, dst, wq, a,
                                                           scores, m_ord, n_edges);

    const int t2 = n_edges * HEADS;
    gat_den_kernel<<<(t2 + 255) / 256, 256, 0, stream>>>(scores, dst, m_ord, den, t2);

    gat_scatter_kernel<<<(t2 + 255) / 256, 256, 0, stream>>>(h, src, dst, scores,
                                                             m_ord, den, out, t2);
}